// RelationModule_73735998538461
// MI455X (gfx1250) — compile-verified
//
#include <hip/hip_runtime.h>
#include <math.h>

#define BB   8
#define KKF  256      // full proposals
#define KV   192      // visible proposals
#define HID  128
#define HEADS 4
#define DKH  32
#define NP   40000
#define NS   1024

typedef __attribute__((ext_vector_type(2))) float v2f;
typedef __attribute__((ext_vector_type(8))) float v8f;

// ---------------------------------------------------------------------------
// Wave-level fp32 WMMA tile: D(16x16) = A(16xkd) * B(kd x16), f32 accumulate.
// Uses V_WMMA_F32_16X16X4_F32 (CDNA5, wave32). Layouts per ISA 7.12.2:
//   A: lane(m)=lane&15 holds row m; VGPR0/1 = K pair, lane halves split K 0..1/2..3
//   C/D: VGPR r -> row (lane>>4)*8 + r, col = lane&15
//   B assumed symmetric to A (K split across lane halves, N striped in lanes)
// BT=true means B is given transposed: Bt[n][k] (row-major, ldb), i.e. we
// compute A * Bt^T. BT=false: B[k][n] row-major (ldb).
// ---------------------------------------------------------------------------
template <bool BT>
__device__ inline v8f wmma_k(const float* __restrict__ A, int lda,
                             const float* __restrict__ B, int ldb, int kd) {
  const int lane = threadIdx.x & 31;
  const int l15  = lane & 15;
  const int kh   = (lane >> 4) << 1;  // 0 or 2
  v8f acc = {0.f, 0.f, 0.f, 0.f, 0.f, 0.f, 0.f, 0.f};
  for (int k0 = 0; k0 < kd; k0 += 4) {
    const int ka = k0 + kh;
    v2f a, b;
    a.x = A[l15 * lda + ka];
    a.y = A[l15 * lda + ka + 1];
    if (BT) {
      b.x = B[l15 * ldb + ka];
      b.y = B[l15 * ldb + ka + 1];
    } else {
      b.x = B[ka * ldb + l15];
      b.y = B[(ka + 1) * ldb + l15];
    }
    acc = __builtin_amdgcn_wmma_f32_16x16x4_f32(false, a, false, b,
                                                (short)0, acc, false, false);
  }
  return acc;
}

__device__ inline void wmma_rc(int& r0, int& c0) {
  const int lane = threadIdx.x & 31;
  r0 = (lane >> 4) * 8;
  c0 = lane & 15;
}

// ---------------------------------------------------------------------------
// conv1 (+BN+PReLU) / conv2 stage. W is (out=128, in=128) row-major -> BT.
// grid.x = nrows/16, block 256 (8 waves = 8 col tiles).
// ---------------------------------------------------------------------------
__global__ void k_conv(const float* __restrict__ X, const float* __restrict__ W,
                       const float* __restrict__ bias,
                       const float* __restrict__ bn_g, const float* __restrict__ bn_b,
                       const float* __restrict__ bn_m, const float* __restrict__ bn_v,
                       const float* __restrict__ prelu,
                       float* __restrict__ Y, int stage) {
  const int rt = blockIdx.x;
  const int wv = threadIdx.x >> 5;
  v8f acc = wmma_k<true>(X + (size_t)rt * 16 * HID, HID,
                         W + (size_t)(wv * 16) * HID, HID, HID);
  int r0, c0; wmma_rc(r0, c0);
  const int n = wv * 16 + c0;
#pragma unroll
  for (int r = 0; r < 8; ++r) {
    float val = acc[r] + bias[n];
    if (stage == 0) {
      val = (val - bn_m[n]) * rsqrtf(bn_v[n] + 1e-5f) * bn_g[n] + bn_b[n];
      val = val > 0.f ? val : prelu[n] * val;
    }
    Y[(size_t)(rt * 16 + r0 + r) * HID + n] = val;
  }
}

// Y = X(nrows x 128) @ W(128 x 128, (din,dout) row-major) + b
__global__ void k_linear(const float* __restrict__ X, const float* __restrict__ W,
                         const float* __restrict__ bias, float* __restrict__ Y) {
  const int rt = blockIdx.x;
  const int wv = threadIdx.x >> 5;
  v8f acc = wmma_k<false>(X + (size_t)rt * 16 * HID, HID, W + wv * 16, HID, HID);
  int r0, c0; wmma_rc(r0, c0);
  const int n = wv * 16 + c0;
#pragma unroll
  for (int r = 0; r < 8; ++r)
    Y[(size_t)(rt * 16 + r0 + r) * HID + n] = acc[r] + bias[n];
}

// features_vis init: first KV rows of each batch
__global__ void k_init_vis(const float* __restrict__ f, float* __restrict__ fv) {
  int i = blockIdx.x * 256 + threadIdx.x;
  if (i >= BB * KV * HID) return;
  int c = i % HID, q = (i / HID) % KV, b = i / (HID * KV);
  fv[i] = f[((size_t)b * KKF + q) * HID + c];
}

// obj_feat irregular gather (replicates the flat-index arithmetic incl. b*128)
__global__ void k_obj_gather(const float* __restrict__ pc, const int* __restrict__ seed,
                             const int* __restrict__ vote, float* __restrict__ obj) {
  int idx = blockIdx.x * 256 + threadIdx.x;
  if (idx >= BB * KKF * HID) return;
  int j  = idx & 127;
  int bk = idx >> 7;
  int b = bk / KKF, k = bk % KKF;
  int s = seed[b * NS + vote[b * KKF + k]];
  int flat = (s + b * 128) * 128 + j;
  int bb2 = flat / (128 * NP);
  int rem = flat % (128 * NP);
  int c = rem / NP;
  int p = rem % NP;
  obj[idx] = pc[((size_t)bb2 * NP + p) * 134 + 6 + c];
}

// manual bbox feature (padded to 28) + corner-mean centers oc
__global__ void k_manual(const float* __restrict__ corner, float* __restrict__ man,
                         float* __restrict__ oc) {
  int bk = blockIdx.x * 256 + threadIdx.x;
  if (bk >= BB * KKF) return;
  const float* c8 = corner + (size_t)bk * 24;
  float mn[3] = {1e30f, 1e30f, 1e30f}, mx[3] = {-1e30f, -1e30f, -1e30f},
        sm[3] = {0.f, 0.f, 0.f};
  for (int i = 0; i < 8; ++i)
    for (int a = 0; a < 3; ++a) {
      float v = c8[i * 3 + a];
      mn[a] = fminf(mn[a], v); mx[a] = fmaxf(mx[a], v); sm[a] += v;
    }
  float* m = man + (size_t)bk * 28;
  float ctr[3];
  for (int a = 0; a < 3; ++a) { ctr[a] = 0.5f * (mn[a] + mx[a]); m[a] = ctr[a]; }
  for (int i = 0; i < 8; ++i)
    for (int a = 0; a < 3; ++a) m[3 + i * 3 + a] = c8[i * 3 + a] - ctr[a];
  m[27] = 0.f;
  for (int a = 0; a < 3; ++a) oc[(size_t)bk * 3 + a] = sm[a] * 0.125f;
}

__global__ void k_pad_wbb(const float* __restrict__ w, float* __restrict__ wp) {
  int i = blockIdx.x * 256 + threadIdx.x;
  if (i >= 28 * HID) return;
  wp[i] = (i < 27 * HID) ? w[i] : 0.f;
}

// per-pair distance MLP: [dx,dy,dz,dist] -> 32 -> 32 -> HEADS, LN after relu
__global__ void k_dist_mlp(const float* __restrict__ oc,
                           const float* __restrict__ l1w, const float* __restrict__ l1b,
                           const float* __restrict__ g1,  const float* __restrict__ b1,
                           const float* __restrict__ l2w, const float* __restrict__ l2b,
                           const float* __restrict__ g2,  const float* __restrict__ b2,
                           const float* __restrict__ l3w, const float* __restrict__ l3b,
                           float* __restrict__ dw) {
  int idx = blockIdx.x * 256 + threadIdx.x;
  if (idx >= BB * KKF * KKF) return;
  int j = idx % KKF, bi = idx / KKF, i = bi % KKF, b = bi / KKF;
  const float* oi = oc + ((size_t)b * KKF + i) * 3;
  const float* oj = oc + ((size_t)b * KKF + j) * 3;
  float w4[4];
  w4[0] = oj[0] - oi[0]; w4[1] = oj[1] - oi[1]; w4[2] = oj[2] - oi[2];
  w4[3] = sqrtf(w4[0] * w4[0] + w4[1] * w4[1] + w4[2] * w4[2]);
  float h[32], h2[32];
#pragma unroll
  for (int n = 0; n < 32; ++n) {
    float s = l1b[n];
    for (int m = 0; m < 4; ++m) s += w4[m] * l1w[m * 32 + n];
    h[n] = s > 0.f ? s : 0.f;
  }
  float mu = 0.f;
  for (int n = 0; n < 32; ++n) mu += h[n];
  mu *= (1.f / 32.f);
  float var = 0.f;
  for (int n = 0; n < 32; ++n) { float d = h[n] - mu; var += d * d; }
  var *= (1.f / 32.f);
  float rs = rsqrtf(var + 1e-5f);
  for (int n = 0; n < 32; ++n) h[n] = (h[n] - mu) * rs * g1[n] + b1[n];
  for (int n = 0; n < 32; ++n) {
    float s = l2b[n];
    for (int m = 0; m < 32; ++m) s += h[m] * l2w[m * 32 + n];
    h2[n] = s > 0.f ? s : 0.f;
  }
  mu = 0.f;
  for (int n = 0; n < 32; ++n) mu += h2[n];
  mu *= (1.f / 32.f);
  var = 0.f;
  for (int n = 0; n < 32; ++n) { float d = h2[n] - mu; var += d * d; }
  var *= (1.f / 32.f);
  rs = rsqrtf(var + 1e-5f);
  for (int n = 0; n < 32; ++n) h2[n] = (h2[n] - mu) * rs * g2[n] + b2[n];
  for (int hh = 0; hh < HEADS; ++hh) {
    float s = l3b[hh];
    for (int m = 0; m < 32; ++m) s += h2[m] * l3w[m * 4 + hh];
    dw[(((size_t)b * HEADS + hh) * KKF + i) * KKF + j] = s;
  }
}

// features += 0.1*(obj@Wobj+bobj) + (man28@Wbb_pad+bbb); mirror into fvis rows
__global__ void k_emb_add(const float* __restrict__ objf, const float* __restrict__ Wobj,
                          const float* __restrict__ bobj, const float* __restrict__ man,
                          const float* __restrict__ Wbb,  const float* __restrict__ bbb,
                          float* __restrict__ feat, float* __restrict__ fvis) {
  const int rt = blockIdx.x;
  const int wv = threadIdx.x >> 5;
  v8f a1 = wmma_k<false>(objf + (size_t)rt * 16 * HID, HID, Wobj + wv * 16, HID, HID);
  v8f a2 = wmma_k<false>(man + (size_t)rt * 16 * 28, 28, Wbb + wv * 16, HID, 28);
  int r0, c0; wmma_rc(r0, c0);
  const int n = wv * 16 + c0;
#pragma unroll
  for (int r = 0; r < 8; ++r) {
    int row = rt * 16 + r0 + r;
    float e = 0.1f * (a1[r] + bobj[n]) + (a2[r] + bbb[n]);
    feat[(size_t)row * HID + n] += e;
    int b = row / KKF, kk = row % KKF;
    if (kk < KV) fvis[((size_t)b * KV + kk) * HID + n] += e;
  }
}

// S[bh,i,j] = scale * <q_h[i], k_h[j]> + dw[bh,i,j]   (dw ld = 256 always)
__global__ void k_scores(const float* __restrict__ q, const float* __restrict__ k,
                         const float* __restrict__ dw, float* __restrict__ S,
                         int N, float scale) {
  const int it = blockIdx.x;
  const int jt = blockIdx.y * 8 + (threadIdx.x >> 5);
  if (jt >= N / 16) return;  // wave-uniform
  const int bh = blockIdx.z, b = bh / HEADS, h = bh % HEADS;
  const float* A  = q + ((size_t)(b * N + it * 16)) * HID + h * DKH;
  const float* Bt = k + ((size_t)(b * N + jt * 16)) * HID + h * DKH;
  v8f acc = wmma_k<true>(A, HID, Bt, HID, DKH);
  int r0, c0; wmma_rc(r0, c0);
#pragma unroll
  for (int r = 0; r < 8; ++r) {
    int i = it * 16 + r0 + r, j = jt * 16 + c0;
    S[((size_t)bh * N + i) * N + j] =
        acc[r] * scale + dw[((size_t)bh * KKF + i) * KKF + j];
  }
}

__global__ void k_softmax(float* __restrict__ S, int N) {
  const int row = blockIdx.x;
  float* p = S + (size_t)row * N;
  __shared__ float sh[256];
  int t = threadIdx.x;
  float v = (t < N) ? p[t] : -1e30f;
  sh[t] = v; __syncthreads();
  for (int s = 128; s > 0; s >>= 1) { if (t < s) sh[t] = fmaxf(sh[t], sh[t + s]); __syncthreads(); }
  float mx = sh[0]; __syncthreads();
  float e = (t < N) ? expf(v - mx) : 0.f;
  sh[t] = e; __syncthreads();
  for (int s = 128; s > 0; s >>= 1) { if (t < s) sh[t] += sh[t + s]; __syncthreads(); }
  float inv = 1.f / sh[0];
  if (t < N) p[t] = e * inv;
}

// out[b,i,h*32 + nt*16 + c] = sum_j P[bh,i,j] * v[b,j,...]
__global__ void k_attout(const float* __restrict__ P, const float* __restrict__ v,
                         float* __restrict__ out, int N) {
  const int it = blockIdx.x * 8 + (threadIdx.x >> 5);
  if (it >= N / 16) return;  // wave-uniform
  const int nt = blockIdx.y;
  const int bh = blockIdx.z, b = bh / HEADS, h = bh % HEADS;
  const float* A  = P + ((size_t)bh * N + it * 16) * N;
  const float* Bp = v + (size_t)b * N * HID + h * DKH + nt * 16;
  v8f acc = wmma_k<false>(A, N, Bp, HID, N);
  int r0, c0; wmma_rc(r0, c0);
#pragma unroll
  for (int r = 0; r < 8; ++r)
    out[((size_t)b * N + it * 16 + r0 + r) * HID + h * DKH + nt * 16 + c0] = acc[r];
}

// y = LayerNorm(x + t) * g + b  (row-wise, 128 wide)
__global__ void k_res_ln(const float* __restrict__ x, const float* __restrict__ t,
                         const float* __restrict__ g, const float* __restrict__ be,
                         float* __restrict__ y) {
  const int row = blockIdx.x, c = threadIdx.x;
  float v = x[(size_t)row * HID + c] + t[(size_t)row * HID + c];
  __shared__ float sh[HID];
  sh[c] = v; __syncthreads();
  for (int s = 64; s > 0; s >>= 1) { if (c < s) sh[c] += sh[c + s]; __syncthreads(); }
  float mean = sh[0] * (1.f / HID); __syncthreads();
  float d = v - mean;
  sh[c] = d * d; __syncthreads();
  for (int s = 64; s > 0; s >>= 1) { if (c < s) sh[c] += sh[c + s]; __syncthreads(); }
  float var = sh[0] * (1.f / HID);
  y[(size_t)row * HID + c] = d * rsqrtf(var + 1e-5f) * g[c] + be[c];
}

// y = p / max(||p||, 1e-12) row-wise
__global__ void k_norm(const float* __restrict__ p, float* __restrict__ y) {
  const int row = blockIdx.x, c = threadIdx.x;
  float v = p[(size_t)row * HID + c];
  __shared__ float sh[HID];
  sh[c] = v * v; __syncthreads();
  for (int s = 64; s > 0; s >>= 1) { if (c < s) sh[c] += sh[c + s]; __syncthreads(); }
  float nrm = fmaxf(sqrtf(sh[0]), 1e-12f);
  y[(size_t)row * HID + c] = v / nrm;
}

// ---------------------------------------------------------------------------
extern "C" void kernel_launch(void* const* d_in, const int* in_sizes, int n_in,
                              void* d_out, int out_size, void* d_ws, size_t ws_size,
                              hipStream_t stream) {
  (void)in_sizes; (void)n_in; (void)out_size; (void)ws_size;
  // --- inputs (dict insertion order; params flattened recursively in order) ---
  const float* pbf    = (const float*)d_in[1];
  const float* corner = (const float*)d_in[2];
  const float* pc     = (const float*)d_in[3];
  const int*   seed   = (const int*)d_in[4];
  const int*   vote   = (const int*)d_in[5];
  auto P = [&](int i) { return (const float*)d_in[i]; };
  const int CONV1_W = 7, CONV1_B = 8, BN_G = 9, BN_B = 10, BN_M = 11, BN_V = 12,
            PRELU = 13, CONV2_W = 14, CONV2_B = 15;
  const int FC0 = 16;    // per depth 10: l1w,l1b,ln1g,ln1b,l2w,l2b,ln2g,ln2b,l3w,l3b
  const int AT0 = 36;    // per depth 10: qw,qb,kw,kb,vw,vb,ow,ob,lng,lnb
  const int BBX0 = 56;   // per depth 2: w,b
  const int OBJ0 = 60;   // per depth 2: w,b
  const int PROJ_W = 64, PROJ_B = 65;

  float* out    = (float*)d_out;
  float* o_fatt = out;                     // (B,K,128)
  float* o_fm   = out + 262144;            // (B,K,128)
  float* o_fvis = out + 524288;            // (B,Kv,128)
  float* o_proj = out + 720896;            // (B,K,128)
  float* dwout  = out + 983040;            // (B,H,K,K) -- also MHA bias buffer

  // --- workspace carve-up (floats) ---
  float* W0 = (float*)d_ws;
  size_t off = 0;
  auto alloc = [&](size_t n) { float* p = W0 + off; off += n; return p; };
  float* htmp  = alloc(2048 * 128);
  float* feat  = alloc(2048 * 128);
  float* fvis  = alloc(1536 * 128);
  float* objf  = alloc(2048 * 128);
  float* man   = alloc(2048 * 28);
  float* oc    = alloc(2048 * 3);
  float* qb    = alloc(2048 * 128);
  float* kb    = alloc(2048 * 128);
  float* vb    = alloc(2048 * 128);
  float* Sb    = alloc((size_t)BB * HEADS * KKF * KKF);
  float* aout  = alloc(2048 * 128);
  float* obuf  = alloc(2048 * 128);
  float* fatt  = alloc(2048 * 128);
  float* projb = alloc(2048 * 128);
  float* wbbp  = alloc(28 * 128);

  const float scale = 1.0f / sqrtf((float)DKH);

  // features_concat
  k_conv<<<128, 256, 0, stream>>>(pbf, P(CONV1_W), P(CONV1_B), P(BN_G), P(BN_B),
                                  P(BN_M), P(BN_V), P(PRELU), htmp, 0);
  k_conv<<<128, 256, 0, stream>>>(htmp, P(CONV2_W), P(CONV2_B), nullptr, nullptr,
                                  nullptr, nullptr, nullptr, feat, 1);
  k_init_vis<<<(BB * KV * HID + 255) / 256, 256, 0, stream>>>(feat, fvis);
  k_obj_gather<<<(BB * KKF * HID + 255) / 256, 256, 0, stream>>>(pc, seed, vote, objf);
  k_manual<<<(BB * KKF + 255) / 256, 256, 0, stream>>>(corner, man, oc);

  for (int d = 0; d < 2; ++d) {
    const int fc = FC0 + d * 10, at = AT0 + d * 10;
    k_pad_wbb<<<14, 256, 0, stream>>>(P(BBX0 + d * 2), wbbp);
    k_dist_mlp<<<(BB * KKF * KKF + 255) / 256, 256, 0, stream>>>(
        oc, P(fc), P(fc + 1), P(fc + 2), P(fc + 3), P(fc + 4), P(fc + 5),
        P(fc + 6), P(fc + 7), P(fc + 8), P(fc + 9), dwout);
    k_emb_add<<<128, 256, 0, stream>>>(objf, P(OBJ0 + d * 2), P(OBJ0 + d * 2 + 1),
                                       man, wbbp, P(BBX0 + d * 2 + 1), feat, fvis);
    if (d == 1) {  // full-proposal MHA: only the last depth's output is live
      k_linear<<<128, 256, 0, stream>>>(feat, P(at + 0), P(at + 1), qb);
      k_linear<<<128, 256, 0, stream>>>(feat, P(at + 2), P(at + 3), kb);
      k_linear<<<128, 256, 0, stream>>>(feat, P(at + 4), P(at + 5), vb);
      k_scores<<<dim3(16, 2, BB * HEADS), 256, 0, stream>>>(qb, kb, dwout, Sb, KKF, scale);
      k_softmax<<<BB * HEADS * KKF, 256, 0, stream>>>(Sb, KKF);
      k_attout<<<dim3(2, 2, BB * HEADS), 256, 0, stream>>>(Sb, vb, aout, KKF);
      k_linear<<<128, 256, 0, stream>>>(aout, P(at + 6), P(at + 7), obuf);
      k_res_ln<<<2048, 128, 0, stream>>>(feat, obuf, P(at + 8), P(at + 9), fatt);
    }
    // visible-proposal MHA (live at both depths; replaces fvis)
    k_linear<<<96, 256, 0, stream>>>(fvis, P(at + 0), P(at + 1), qb);
    k_linear<<<96, 256, 0, stream>>>(fvis, P(at + 2), P(at + 3), kb);
    k_linear<<<96, 256, 0, stream>>>(fvis, P(at + 4), P(at + 5), vb);
    k_scores<<<dim3(12, 2, BB * HEADS), 256, 0, stream>>>(qb, kb, dwout, Sb, KV, scale);
    k_softmax<<<BB * HEADS * KV, 256, 0, stream>>>(Sb, KV);
    k_attout<<<dim3(2, 2, BB * HEADS), 256, 0, stream>>>(Sb, vb, aout, KV);
    k_linear<<<96, 256, 0, stream>>>(aout, P(at + 6), P(at + 7), obuf);
    k_res_ln<<<1536, 128, 0, stream>>>(fvis, obuf, P(at + 8), P(at + 9), fvis);
  }

  hipMemcpyAsync(o_fatt, fatt, 2048 * 128 * sizeof(float), hipMemcpyDeviceToDevice, stream);
  hipMemcpyAsync(o_fm,   fatt, 2048 * 128 * sizeof(float), hipMemcpyDeviceToDevice, stream);
  hipMemcpyAsync(o_fvis, fvis, 1536 * 128 * sizeof(float), hipMemcpyDeviceToDevice, stream);
  k_linear<<<128, 256, 0, stream>>>(fatt, P(PROJ_W), P(PROJ_B), projb);
  k_norm<<<2048, 128, 0, stream>>>(projb, o_proj);
  // dist_weights_out already written in-place by depth-1 k_dist_mlp
}